// ExpertDistillationLoss_17102559773158
// MI455X (gfx1250) — compile-verified
//
#include <hip/hip_runtime.h>
#include <stdint.h>

typedef __attribute__((ext_vector_type(2))) float v2f;
typedef __attribute__((ext_vector_type(8))) float v8f;

#define ET 128
#define ES 16
#define KSTEPS 3
#define ALPHA_F 0.5f
#define LAMBDA_COV_F 0.5f
#define EPS_F 1e-8f

// ws float layout: [0]=sum_kw [1]=sum_w [2,3]=pad (16B align) [4..131]=t_counts [132..259]=s_counts
#define WS_FLOATS 260

__device__ __forceinline__ unsigned rotl32(unsigned x, unsigned n) {
  return (x << n) | (x >> (32u - n));
}

// Counter-based deterministic RNG (threefry2x32-style, 20 rounds), key=(0,42).
__device__ __forceinline__ unsigned threefry2x32_20(unsigned c0, unsigned c1) {
  const unsigned k0 = 0u, k1 = 42u;
  const unsigned ks2 = 0x1BD11BDAu ^ k0 ^ k1;
  const unsigned ks[3] = {k0, k1, ks2};
  const unsigned R[8] = {13u, 15u, 26u, 6u, 17u, 29u, 16u, 24u};
  unsigned x0 = c0 + k0, x1 = c1 + k1;
#pragma unroll
  for (int g = 0; g < 5; ++g) {
#pragma unroll
    for (int r = 0; r < 4; ++r) {
      x0 += x1;
      x1 = rotl32(x1, R[(g & 1) * 4 + r]);
      x1 ^= x0;
    }
    x0 += ks[(g + 1) % 3];
    x1 += ks[(g + 2) % 3] + (unsigned)(g + 1);
  }
  return x0;
}

__global__ void k_init(float* __restrict__ acc) {
  for (int i = threadIdx.x; i < WS_FLOATS; i += blockDim.x) acc[i] = 0.f;
}

// One token per wave32. Lane l holds teacher experts [4l..4l+3] in VGPRs across
// all K MC steps. Student-gate row is async-staged to LDS (ASYNCcnt path).
__global__ void __launch_bounds__(256) k_main(const float* __restrict__ tg,
                                              const float* __restrict__ sg,
                                              const int* __restrict__ am,
                                              float* __restrict__ acc, int Ntok) {
  __shared__ float s_t[ET];
  __shared__ float s_s[ET];
  __shared__ float s_scalar[2];  // [0]=kw, [1]=w
  __shared__ float stu[8 * ES];  // per-wave student-gate row staging

  const int tid = threadIdx.x;
  for (int i = tid; i < ET; i += blockDim.x) { s_t[i] = 0.f; s_s[i] = 0.f; }
  if (tid < 2) s_scalar[tid] = 0.f;
  __syncthreads();

  const int lane = tid & 31;
  const int wv = tid >> 5;
  const int gwave = blockIdx.x * (blockDim.x >> 5) + wv;
  const int nwaves = gridDim.x * (blockDim.x >> 5);

  const float4* __restrict__ tg4 = (const float4*)tg;
  // Generic pointers to LDS carry the LDS byte offset in the low 32 bits.
  const unsigned stu_lds = (unsigned)(uintptr_t)&stu[wv * ES + (lane & 15)];

  float acc_kw = 0.f, acc_w = 0.f;

  for (int t = gwave; t < Ntok; t += nwaves) {
    // Async-stage this token's 16-float student row into LDS (lanes 16..31
    // duplicate lanes 0..15: identical data to identical LDS addresses).
    {
      unsigned long long sbase = (unsigned long long)(uintptr_t)(sg + (size_t)t * ES);
      unsigned voff = (unsigned)((lane & 15) * 4);
      asm volatile("global_load_async_to_lds_b32 %0, %1, %2 offset:0"
                   :
                   : "v"(stu_lds), "v"(voff), "s"(sbase)
                   : "memory");
    }

    // Coalesced b128 teacher load: 4 experts per lane, kept in VGPRs for all K steps.
    float4 pv = tg4[(size_t)t * (ET / 4) + lane];
    float p0 = pv.x, p1 = pv.y, p2 = pv.z, p3 = pv.w;
    float vm = (am[t] != 0) ? 1.f : 0.f;

    int tn = t + nwaves;
    if (tn < Ntok) __builtin_prefetch(&tg4[(size_t)tn * (ET / 4) + lane], 0, 1);

    asm volatile("s_wait_asynccnt 0" ::: "memory");

#pragma unroll
    for (int k = 0; k < KSTEPS; ++k) {
      // Wave-wide inclusive prefix sum over 32 lane-partials (inverse-CDF sampling).
      float lsum = p0 + p1 + p2 + p3;
      float incl = lsum;
#pragma unroll
      for (int d = 1; d < 32; d <<= 1) {
        float y = __shfl_up(incl, d, 32);
        if (lane >= d) incl += y;
      }
      float excl = incl - lsum;
      float total = __shfl(incl, 31, 32);

      float u = (float)(threefry2x32_20((unsigned)t, (unsigned)k) >> 8) * 0x1.0p-24f;
      float target = u * total;

      bool own = (target >= excl) && (target < incl);
      unsigned mb = (unsigned)__ballot(own);
      int ol = mb ? (__ffs(mb) - 1) : 31;

      int jloc;
      float wq;
      float c0 = excl + p0, c1a = c0 + p1, c2a = c1a + p2;
      if (target < c0) { jloc = 0; wq = p0; }
      else if (target < c1a) { jloc = 1; wq = p1; }
      else if (target < c2a) { jloc = 2; wq = p2; }
      else { jloc = 3; wq = p3; }

      int idx = __shfl(lane * 4 + jloc, ol, 32);
      float wun = __shfl(wq, ol, 32);
      float w = wun / total;  // normalized prob at sampling time == importance weight

      // Damp the sampled expert (implicit renormalization via recomputed total).
      if (lane == ol) {
        if (jloc == 0) p0 *= ALPHA_F;
        else if (jloc == 1) p1 *= ALPHA_F;
        else if (jloc == 2) p2 *= ALPHA_F;
        else p3 *= ALPHA_F;
      }

      if (lane == 0) {
        float wsv = w * vm;
        acc_w += wsv;
        if ((idx & 7) == 0)  // hit: teacher idx maps onto a student expert
          acc_kw += (-logf(stu[wv * ES + (idx >> 3)])) * wsv;
        if (idx < ES) {  // in_range coverage counts
          atomicAdd(&s_t[idx], wsv);
          atomicAdd(&s_s[idx], stu[wv * ES + idx] * vm);
        }
      }
    }
  }

  if (lane == 0) {
    atomicAdd(&s_scalar[0], acc_kw);
    atomicAdd(&s_scalar[1], acc_w);
  }
  __syncthreads();
  for (int i = tid; i < ET; i += blockDim.x) {
    atomicAdd(&acc[4 + i], s_t[i]);
    atomicAdd(&acc[132 + i], s_s[i]);
  }
  if (tid == 0) atomicAdd(&acc[0], s_scalar[0]);
  if (tid == 1) atomicAdd(&acc[1], s_scalar[1]);
}

// Single-wave finalize: coverage KL's 128-element dot product via two
// V_WMMA_F32_16X16X4_F32 (trace trick: A[m][k]=x[4m+k], B[k][n]=y[4n+k]
// => trace(A*B) = dot(x,y)), exact f32.
__global__ void __launch_bounds__(32) k_final(const float* __restrict__ acc,
                                              float* __restrict__ out) {
  const int lane = threadIdx.x;
  float4 tcv = ((const float4*)(acc + 4))[lane];
  float4 scv = ((const float4*)(acc + 132))[lane];

  float st = tcv.x + tcv.y + tcv.z + tcv.w;
  float ss = scv.x + scv.y + scv.z + scv.w;
#pragma unroll
  for (int d = 16; d > 0; d >>= 1) {
    st += __shfl_xor(st, d, 32);
    ss += __shfl_xor(ss, d, 32);
  }

  float4 ta, sa;
  ta.x = tcv.x / st + EPS_F; ta.y = tcv.y / st + EPS_F;
  ta.z = tcv.z / st + EPS_F; ta.w = tcv.w / st + EPS_F;
  sa.x = scv.x / ss + EPS_F; sa.y = scv.y / ss + EPS_F;
  sa.z = scv.z / ss + EPS_F; sa.w = scv.w / ss + EPS_F;

  float st2 = ta.x + ta.y + ta.z + ta.w;
  float ss2 = sa.x + sa.y + sa.z + sa.w;
#pragma unroll
  for (int d = 16; d > 0; d >>= 1) {
    st2 += __shfl_xor(st2, d, 32);
    ss2 += __shfl_xor(ss2, d, 32);
  }
  ta.x /= st2; ta.y /= st2; ta.z /= st2; ta.w /= st2;
  sa.x /= ss2; sa.y /= ss2; sa.z /= ss2; sa.w /= ss2;

  float yx = logf(ta.x) - logf(sa.x);
  float yy = logf(ta.y) - logf(sa.y);
  float yz = logf(ta.z) - logf(sa.z);
  float yw = logf(ta.w) - logf(sa.w);

  v8f c = {0.f, 0.f, 0.f, 0.f, 0.f, 0.f, 0.f, 0.f};
#pragma unroll
  for (int h = 0; h < 2; ++h) {
    int sl = h * 16 + (lane & 15);
    float a0 = __shfl(ta.x, sl, 32), a1 = __shfl(ta.y, sl, 32);
    float a2 = __shfl(ta.z, sl, 32), a3 = __shfl(ta.w, sl, 32);
    float b0 = __shfl(yx, sl, 32), b1 = __shfl(yy, sl, 32);
    float b2 = __shfl(yz, sl, 32), b3 = __shfl(yw, sl, 32);
    // A 16x4 f32 layout: lanes 0-15 -> M=lane, K=0/1; lanes 16-31 -> M=lane-16, K=2/3.
    v2f A; A.x = (lane < 16) ? a0 : a2; A.y = (lane < 16) ? a1 : a3;
    v2f B; B.x = (lane < 16) ? b0 : b2; B.y = (lane < 16) ? b1 : b3;
    c = __builtin_amdgcn_wmma_f32_16x16x4_f32(false, A, false, B, (short)0, c,
                                              false, false);
  }

  float carr[8];
#pragma unroll
  for (int i = 0; i < 8; ++i) carr[i] = c[i];
  // Diagonal of 16x16 f32 C/D: i<8 -> vgpr i lane i; i>=8 -> vgpr i-8 lane i+16.
  float diag = 0.f;
  if (lane < 8) diag = carr[lane];
  else if (lane >= 24) diag = carr[lane - 24];
#pragma unroll
  for (int d = 16; d > 0; d >>= 1) diag += __shfl_xor(diag, d, 32);

  float cov = diag / (float)ET;
  float feat = acc[0] / fmaxf(acc[1], EPS_F);
  if (lane == 0) out[0] = feat + LAMBDA_COV_F * cov;
}

extern "C" void kernel_launch(void* const* d_in, const int* in_sizes, int n_in,
                              void* d_out, int out_size, void* d_ws, size_t ws_size,
                              hipStream_t stream) {
  (void)n_in; (void)out_size; (void)ws_size;
  const float* tg = (const float*)d_in[0];  // teacher_gates  [N, 128]
  const float* sg = (const float*)d_in[1];  // student_gates  [N, 16]
  const int* am = (const int*)d_in[4];      // attention_mask [N]
  // d_in[2], d_in[3] (hidden states) are dead inputs for this loss: never read.
  float* out = (float*)d_out;
  float* acc = (float*)d_ws;
  const int N = in_sizes[4];  // B*S tokens

  k_init<<<dim3(1), dim3(256), 0, stream>>>(acc);
  k_main<<<dim3(1024), dim3(256), 0, stream>>>(tg, sg, am, acc, N);
  k_final<<<dim3(1), dim3(32), 0, stream>>>(acc, out);
}